// Model1_52518860096440
// MI455X (gfx1250) — compile-verified
//
#include <hip/hip_runtime.h>
#include <math.h>
#include <stdint.h>

// ---------------------------------------------------------------------------
// MI455X (gfx1250): fused-QKV projection + per-position 16x16 head attention.
//
//   qkv = query @ W_qkv^T + b   (GEMM M=16384, N=3072, K=1024, fp32)
//   per position: scores = q k^T/8 + mask; softmax over 16; o = P v
//
// Preferred path (192MB scratch): blocked 128x128 GEMM -> ws, then attention
// kernel staging 16 qkv rows in LDS.  Tile staging uses the CDNA5 Tensor Data
// Mover (tensor_load_to_lds, TENSORcnt) with double-buffered LDS: the DMA
// engine streams A/W tiles (with native pad-to-stride-36 for bank-conflict-
// free WMMA fragment reads) while the waves run pure WMMA out of LDS.
// Fabric traffic ~3.2GB vs ~12.9GB for the fused variant.
//
// Fallback path (no scratch): single fused kernel, 16x3072 qkv tile in LDS.
// All matrix math uses V_WMMA_F32_16X16X4_F32 (exact fp32 tensor path).
// ---------------------------------------------------------------------------

typedef __attribute__((ext_vector_type(2))) float v2f;
typedef __attribute__((ext_vector_type(8))) float v8f;

#define B_    4
#define S_    4096
#define HID_  1024
#define H_    16
#define DH_   64
#define N3_   3072      // 3*HID
#define K_    1024
#define M_    (B_ * S_) // 16384

// ====================== Tensor Data Mover (TDM) helpers =====================
#if defined(__has_builtin)
#if __has_builtin(__builtin_amdgcn_tensor_load_to_lds)
#define HAVE_TDM 1
#endif
#endif

#ifdef HAVE_TDM
typedef unsigned int tdm_v4u __attribute__((ext_vector_type(4)));
typedef int          tdm_v8i __attribute__((ext_vector_type(8)));
typedef int          tdm_v4i __attribute__((ext_vector_type(4)));

__device__ __forceinline__ void tdm_issue(tdm_v4u g0, tdm_v8i g1, tdm_v4i g2, tdm_v4i g3) {
#if __clang_major__ >= 23
    tdm_v8i gx = (tdm_v8i){0, 0, 0, 0, 0, 0, 0, 0};
    __builtin_amdgcn_tensor_load_to_lds(g0, g1, g2, g3, gx, 0);
#else
    __builtin_amdgcn_tensor_load_to_lds(g0, g1, g2, g3, 0);
#endif
}

// D# group0: count=1(user), lds byte addr, 57-bit global addr, type=2("image")
__device__ __forceinline__ tdm_v4u tdm_g0(uint32_t lds_byte_off, const void* gsrc) {
    const uint64_t ga = (uint64_t)(uintptr_t)gsrc;
    tdm_v4u g0;
    g0.x = 1u;
    g0.y = lds_byte_off;
    g0.z = (uint32_t)ga;
    g0.w = ((uint32_t)(ga >> 32) & 0x01FFFFFFu) | (2u << 30);
    return g0;
}

// 2D fp32 tile load: tile_x dwords per row, tile_y rows, row stride (elems);
// optional LDS padding (pad_amt_code+1 dwords every (2<<pad_int_code) dwords).
__device__ __forceinline__ void tdm_load_2d_f32(uint32_t lds_byte_off, const float* gsrc,
                                                uint32_t tile_x, uint32_t tile_y,
                                                uint32_t row_stride_elems,
                                                uint32_t pad_en, uint32_t pad_int_code,
                                                uint32_t pad_amt_code) {
    tdm_v8i g1;
    g1[0] = (int)((2u << 16) | (pad_en << 20) | (pad_int_code << 22) | (pad_amt_code << 25));
    g1[1] = (int)(0xFFFFu << 16);                    // tensor_dim0 lo (huge -> no OOB)
    g1[2] = (int)(0x7FFFu | (0xFFFFu << 16));        // tensor_dim0 hi | tensor_dim1 lo
    g1[3] = (int)(0x7FFFu | (tile_x << 16));         // tensor_dim1 hi | tile_dim0
    g1[4] = (int)(tile_y & 0xFFFFu);                 // tile_dim1 (tile_dim2=0)
    g1[5] = (int)row_stride_elems;                   // tensor_dim0_stride lo
    g1[6] = 0;                                       // stride0 hi | stride1 lo
    g1[7] = 0;
    tdm_v4i gz = (tdm_v4i){0, 0, 0, 0};
    tdm_issue(tdm_g0(lds_byte_off, gsrc), g1, gz, gz);
}

// Iterated row load: nrows rows of row_elems fp32; per iteration the global
// address advances gstep elems and the LDS address advances lstep elems.
__device__ __forceinline__ void tdm_load_rows_f32(uint32_t lds_byte_off, const float* gsrc,
                                                  uint32_t row_elems, uint32_t nrows,
                                                  uint32_t gstep_elems, uint32_t lstep_elems) {
    tdm_v8i g1;
    g1[0] = (int)((2u << 16) | (1u << 19));          // data_size=4B, iterate_enable
    g1[1] = (int)((row_elems & 0xFFFFu) << 16);      // tensor_dim0 lo
    g1[2] = (int)(((row_elems >> 16) & 0xFFFFu) | (0xFFFFu << 16)); // dim0 hi | dim1 lo
    g1[3] = (int)(0x7FFFu | (row_elems << 16));      // dim1 hi | tile_dim0
    g1[4] = 0;                                       // tile_dim1=0 (1D tile)
    g1[5] = (int)row_elems;                          // stride0
    g1[6] = 0;
    g1[7] = 0;
    tdm_v4i g2;
    g2[0] = 0;                                       // tensor_dim2 (unused)
    g2[1] = (int)lstep_elems;                        // lds_addr_increment
    g2[2] = (int)gstep_elems;                        // global_addr_increment lo
    g2[3] = (int)((nrows - 1u) << 16);               // inc hi=0 | iterate_count
    tdm_v4i g3 = (tdm_v4i){0, 0, 0, 0};
    tdm_issue(tdm_g0(lds_byte_off, gsrc), g1, g2, g3);
}
#endif  // HAVE_TDM

// ========================== Kernel 1: blocked GEMM ==========================
// C[M,N3] = A[M,K] * W^T + b, 128x128 tile per WG, BK=32, 8 wave32,
// wave grid 2(M) x 4(N), 4x2 WMMA frags per wave, double-buffered TDM staging.
#define TM  128
#define TN  128
#define BK  32
#define LDT 36          // LDS tile row stride (floats): float4-aligned, conflict-free

__launch_bounds__(256)
__global__ void qkv_gemm(const float* __restrict__ A,     // (16384,1024)
                         const float* __restrict__ W,     // (3072,1024) row-major
                         const float* __restrict__ bias,  // (3072,)
                         float* __restrict__ qkv)         // (16384,3072) ws
{
    __shared__ float smem[2 * (TM + TN) * LDT];           // 73,728 B, 2 buffers

    const int tid  = threadIdx.x;
    const int lane = tid & 31;
    const int wave = tid >> 5;
    const int lo16 = lane & 15;
    const int kh   = lane >> 4;
    const int mw   = wave & 1;
    const int nw   = wave >> 1;
    const int r0   = blockIdx.y * TM;
    const int n0   = blockIdx.x * TN;
    const int NCH  = K_ / BK;

    v8f acc[4][2];
    #pragma unroll
    for (int fm = 0; fm < 4; ++fm)
        #pragma unroll
        for (int fn = 0; fn < 2; ++fn)
            acc[fm][fn] = (v8f){0.f,0.f,0.f,0.f,0.f,0.f,0.f,0.f};

#ifdef HAVE_TDM
    // flat->LDS mapping keeps the wave-relative offset in addr[31:0]
    const uint32_t lds0 = (uint32_t)(uintptr_t)(void*)smem;
    // pad codes: interval 32 dwords (code 4), amount 4 dwords (code 3) -> stride 36
    if (wave == 0) {
        tdm_load_2d_f32(lds0, A + (size_t)r0 * K_, BK, TM, K_, 1u, 4u, 3u);
        tdm_load_2d_f32(lds0 + TM * LDT * 4, W + (size_t)n0 * K_, BK, TN, K_, 1u, 4u, 3u);
    }
#else
    const int srow = tid >> 3;        // staging map: 4 float4 per thread
    const int scol = (tid & 7) * 4;
    float4 ra[4], rw[4];
    #pragma unroll
    for (int p = 0; p < 4; ++p) {
        ra[p] = *(const float4*)(A + (size_t)(r0 + srow + 32 * p) * K_ + scol);
        rw[p] = *(const float4*)(W + (size_t)(n0 + srow + 32 * p) * K_ + scol);
    }
#endif

    for (int kc = 0; kc < NCH; ++kc) {
        const int buf = kc & 1;
        float* Asb = smem + buf * (TM + TN) * LDT;
        float* Wsb = Asb + TM * LDT;

#ifdef HAVE_TDM
        if (wave == 0) __builtin_amdgcn_s_wait_tensorcnt(0);   // chunk kc landed
        __syncthreads();
        if (wave == 0 && kc + 1 < NCH) {                        // DMA chunk kc+1
            const uint32_t off = lds0 + (uint32_t)((buf ^ 1) * (TM + TN) * LDT * 4);
            const int k0n = (kc + 1) * BK;
            tdm_load_2d_f32(off, A + (size_t)r0 * K_ + k0n, BK, TM, K_, 1u, 4u, 3u);
            tdm_load_2d_f32(off + TM * LDT * 4, W + (size_t)n0 * K_ + k0n, BK, TN, K_, 1u, 4u, 3u);
        }
#else
        #pragma unroll
        for (int p = 0; p < 4; ++p) {
            *(float4*)(&Asb[(srow + 32 * p) * LDT + scol]) = ra[p];
            *(float4*)(&Wsb[(srow + 32 * p) * LDT + scol]) = rw[p];
        }
        __syncthreads();
        if (kc + 1 < NCH) {
            const int k0n = (kc + 1) * BK;
            #pragma unroll
            for (int p = 0; p < 4; ++p) {
                ra[p] = *(const float4*)(A + (size_t)(r0 + srow + 32 * p) * K_ + k0n + scol);
                rw[p] = *(const float4*)(W + (size_t)(n0 + srow + 32 * p) * K_ + k0n + scol);
            }
        }
#endif

        // 8 K-steps of 4; 8 independent WMMA chains per wave
        #pragma unroll
        for (int ks = 0; ks < 8; ++ks) {
            const int kk = ks * 4 + 2 * kh;
            v2f af[4], bf[2];
            #pragma unroll
            for (int fm = 0; fm < 4; ++fm)
                af[fm] = *(const v2f*)(&Asb[(mw * 64 + fm * 16 + lo16) * LDT + kk]);
            #pragma unroll
            for (int fn = 0; fn < 2; ++fn)
                bf[fn] = *(const v2f*)(&Wsb[(nw * 32 + fn * 16 + lo16) * LDT + kk]);
            #pragma unroll
            for (int fm = 0; fm < 4; ++fm)
                #pragma unroll
                for (int fn = 0; fn < 2; ++fn)
                    acc[fm][fn] = __builtin_amdgcn_wmma_f32_16x16x4_f32(
                        false, af[fm], false, bf[fn], (short)0, acc[fm][fn], false, false);
        }
        __syncthreads();
    }

    // epilogue: + bias, store to workspace
    #pragma unroll
    for (int fn = 0; fn < 2; ++fn) {
        const int ncol = n0 + nw * 32 + fn * 16 + lo16;
        const float bv = bias[ncol];
        #pragma unroll
        for (int fm = 0; fm < 4; ++fm)
            #pragma unroll
            for (int j = 0; j < 8; ++j) {
                const int row = r0 + mw * 64 + fm * 16 + j + 8 * kh;
                qkv[(size_t)row * N3_ + ncol] = acc[fm][fn][j] + bv;
            }
    }
}

// ===================== Kernel 2: per-position attention =====================
#define LDQ 3076        // padded qkv LDS row stride (floats)

__launch_bounds__(256)
__global__ void attn16(const float* __restrict__ qkv,      // (16384,3072) ws
                       const float* __restrict__ attn_mask,
                       float* __restrict__ out)            // (16384,1024)
{
    __shared__ float QKVs[16 * LDQ];   // 196,864 B
    __shared__ float SC[8 * 256];      //   8,192 B per-wave softmax scratch

    const int tid  = threadIdx.x;
    const int lane = tid & 31;
    const int wave = tid >> 5;
    const int lo16 = lane & 15;
    const int kh   = lane >> 4;
    const int r0   = blockIdx.x * 16;

#ifdef HAVE_TDM
    // one TDM iterated-row load stages the whole 16x3072 tile (LDS stride 3076)
    if (wave == 0) {
        tdm_load_rows_f32((uint32_t)(uintptr_t)(void*)QKVs,
                          qkv + (size_t)r0 * N3_, N3_, 16u, N3_, LDQ);
        __builtin_amdgcn_s_wait_tensorcnt(0);
    }
    __syncthreads();
#else
    #pragma unroll
    for (int i = 0; i < 16; ++i)
        #pragma unroll
        for (int j = 0; j < 3; ++j) {
            const int c = (j * 256 + tid) * 4;
            float4 v = *(const float4*)(qkv + (size_t)(r0 + i) * N3_ + c);
            *(float4*)(&QKVs[i * LDQ + c]) = v;
        }
    __syncthreads();
#endif

    const float scale = 0.125f;        // 1/sqrt(64)
    float* sc = &SC[wave * 256];

    for (int pp = 0; pp < 2; ++pp) {
        const int m  = wave * 2 + pp;
        const int gm = r0 + m;
        const float* qrow = &QKVs[m * LDQ];

        // scores = q k^T : 16 chained f32 WMMAs over K=64
        v8f acc = {0.f,0.f,0.f,0.f,0.f,0.f,0.f,0.f};
        const float* qa = qrow + lo16 * DH_ + 2 * kh;
        const float* kb = qrow + HID_ + lo16 * DH_ + 2 * kh;
        #pragma unroll
        for (int k0 = 0; k0 < DH_; k0 += 4) {
            v2f a = *(const v2f*)(qa + k0);
            v2f b = *(const v2f*)(kb + k0);
            acc = __builtin_amdgcn_wmma_f32_16x16x4_f32(false, a, false, b,
                                                        (short)0, acc, false, false);
        }
        const float* mrow = attn_mask + (size_t)gm * (H_ * H_);
        #pragma unroll
        for (int j = 0; j < 8; ++j) {
            const int h = j + 8 * kh;
            sc[h * 16 + lo16] = acc[j] * scale + mrow[h * 16 + lo16];
        }
        __syncthreads();

        if (lane < 16) {               // row-wise softmax over 16 cols
            float mx = -INFINITY;
            #pragma unroll
            for (int g = 0; g < 16; ++g) mx = fmaxf(mx, sc[lane * 16 + g]);
            float e[16]; float sum = 0.f;
            #pragma unroll
            for (int g = 0; g < 16; ++g) { e[g] = __expf(sc[lane * 16 + g] - mx); sum += e[g]; }
            const float inv = 1.0f / sum;
            #pragma unroll
            for (int g = 0; g < 16; ++g) sc[lane * 16 + g] = e[g] * inv;
        }
        __syncthreads();

        // o = P @ V : four 16x16x16 tiles
        float* orow = out + (size_t)gm * HID_;
        #pragma unroll
        for (int d0 = 0; d0 < DH_; d0 += 16) {
            v8f oacc = {0.f,0.f,0.f,0.f,0.f,0.f,0.f,0.f};
            #pragma unroll
            for (int k0 = 0; k0 < 16; k0 += 4) {
                const int kk = k0 + 2 * kh;
                v2f a = *(const v2f*)(sc + lo16 * 16 + kk);
                v2f b;
                b.x = qrow[2 * HID_ + kk * DH_ + d0 + lo16];
                b.y = qrow[2 * HID_ + (kk + 1) * DH_ + d0 + lo16];
                oacc = __builtin_amdgcn_wmma_f32_16x16x4_f32(false, a, false, b,
                                                             (short)0, oacc, false, false);
            }
            #pragma unroll
            for (int j = 0; j < 8; ++j)
                orow[(j + 8 * kh) * DH_ + d0 + lo16] = oacc[j];
        }
    }
}

// ================= Fallback: fully fused kernel (no scratch) ================
#define LDA 1028

__launch_bounds__(256)
__global__ void fused_qkv_attn(const float* __restrict__ q_in,
                               const float* __restrict__ attn_mask,
                               const float* __restrict__ W,
                               const float* __restrict__ bias,
                               float* __restrict__ out)
{
    __shared__ float As[16 * LDA];    //  65,792 B (A tile; softmax scratch in phase 2)
    __shared__ float QKVs[16 * LDQ];  // 196,864 B

    const int tid  = threadIdx.x;
    const int lane = tid & 31;
    const int wave = tid >> 5;
    const int lo16 = lane & 15;
    const int kh   = lane >> 4;
    const int r0   = blockIdx.x * 16;

    #pragma unroll
    for (int i = 0; i < 16; ++i) {
        float4 v = *(const float4*)(q_in + (size_t)(r0 + i) * K_ + tid * 4);
        *(float4*)(&As[i * LDA + tid * 4]) = v;
    }
    __syncthreads();

    const float* Afrag = &As[lo16 * LDA + 2 * kh];
    for (int tt = 0; tt < 8; ++tt) {
        const int t0  = wave * 24 + tt * 3;
        const int n0a = (t0 + 0) * 16, n0b = (t0 + 1) * 16, n0c = (t0 + 2) * 16;
        const float ba = bias[n0a + lo16];
        const float bb = bias[n0b + lo16];
        const float bc = bias[n0c + lo16];
        const float* Wa = W + (size_t)(n0a + lo16) * K_ + 2 * kh;
        const float* Wb = W + (size_t)(n0b + lo16) * K_ + 2 * kh;
        const float* Wc = W + (size_t)(n0c + lo16) * K_ + 2 * kh;
        v8f acca = {0.f,0.f,0.f,0.f,0.f,0.f,0.f,0.f};
        v8f accb = {0.f,0.f,0.f,0.f,0.f,0.f,0.f,0.f};
        v8f accc = {0.f,0.f,0.f,0.f,0.f,0.f,0.f,0.f};
        #pragma unroll 4
        for (int k0 = 0; k0 < K_; k0 += 4) {
            v2f a  = *(const v2f*)(Afrag + k0);
            v2f wa = *(const v2f*)(Wa + k0);
            v2f wb = *(const v2f*)(Wb + k0);
            v2f wc = *(const v2f*)(Wc + k0);
            acca = __builtin_amdgcn_wmma_f32_16x16x4_f32(false, a, false, wa, (short)0, acca, false, false);
            accb = __builtin_amdgcn_wmma_f32_16x16x4_f32(false, a, false, wb, (short)0, accb, false, false);
            accc = __builtin_amdgcn_wmma_f32_16x16x4_f32(false, a, false, wc, (short)0, accc, false, false);
        }
        #pragma unroll
        for (int j = 0; j < 8; ++j) {
            const int m = j + 8 * kh;
            QKVs[m * LDQ + n0a + lo16] = acca[j] + ba;
            QKVs[m * LDQ + n0b + lo16] = accb[j] + bb;
            QKVs[m * LDQ + n0c + lo16] = accc[j] + bc;
        }
    }
    __syncthreads();

    const float scale = 0.125f;
    float* sc = &As[wave * 256];

    for (int pp = 0; pp < 2; ++pp) {
        const int m  = wave * 2 + pp;
        const int gm = r0 + m;
        const float* qrow = &QKVs[m * LDQ];

        v8f acc = {0.f,0.f,0.f,0.f,0.f,0.f,0.f,0.f};
        const float* qa = qrow + lo16 * DH_ + 2 * kh;
        const float* kb = qrow + HID_ + lo16 * DH_ + 2 * kh;
        #pragma unroll
        for (int k0 = 0; k0 < DH_; k0 += 4) {
            v2f a = *(const v2f*)(qa + k0);
            v2f b = *(const v2f*)(kb + k0);
            acc = __builtin_amdgcn_wmma_f32_16x16x4_f32(false, a, false, b,
                                                        (short)0, acc, false, false);
        }
        const float* mrow = attn_mask + (size_t)gm * (H_ * H_);
        #pragma unroll
        for (int j = 0; j < 8; ++j) {
            const int h = j + 8 * kh;
            sc[h * 16 + lo16] = acc[j] * scale + mrow[h * 16 + lo16];
        }
        __syncthreads();

        if (lane < 16) {
            float mx = -INFINITY;
            #pragma unroll
            for (int g = 0; g < 16; ++g) mx = fmaxf(mx, sc[lane * 16 + g]);
            float e[16]; float sum = 0.f;
            #pragma unroll
            for (int g = 0; g < 16; ++g) { e[g] = __expf(sc[lane * 16 + g] - mx); sum += e[g]; }
            const float inv = 1.0f / sum;
            #pragma unroll
            for (int g = 0; g < 16; ++g) sc[lane * 16 + g] = e[g] * inv;
        }
        __syncthreads();

        float* orow = out + (size_t)gm * HID_;
        #pragma unroll
        for (int d0 = 0; d0 < DH_; d0 += 16) {
            v8f oacc = {0.f,0.f,0.f,0.f,0.f,0.f,0.f,0.f};
            #pragma unroll
            for (int k0 = 0; k0 < 16; k0 += 4) {
                const int kk = k0 + 2 * kh;
                v2f a = *(const v2f*)(sc + lo16 * 16 + kk);
                v2f b;
                b.x = qrow[2 * HID_ + kk * DH_ + d0 + lo16];
                b.y = qrow[2 * HID_ + (kk + 1) * DH_ + d0 + lo16];
                oacc = __builtin_amdgcn_wmma_f32_16x16x4_f32(false, a, false, b,
                                                             (short)0, oacc, false, false);
            }
            #pragma unroll
            for (int j = 0; j < 8; ++j)
                orow[(j + 8 * kh) * DH_ + d0 + lo16] = oacc[j];
        }
    }
}

// =============================== dispatch ===================================
extern "C" void kernel_launch(void* const* d_in, const int* in_sizes, int n_in,
                              void* d_out, int out_size, void* d_ws, size_t ws_size,
                              hipStream_t stream) {
    (void)in_sizes; (void)n_in; (void)out_size;
    const float* query = (const float*)d_in[0];
    // d_in[1] = key, d_in[2] = value : unused by the reference computation
    const float* mask  = (const float*)d_in[3];
    const float* W     = (const float*)d_in[4];
    const float* bias  = (const float*)d_in[5];
    float* out = (float*)d_out;

    const size_t need = (size_t)M_ * N3_ * sizeof(float);   // 192 MB qkv scratch
    if (d_ws != nullptr && ws_size >= need) {
        float* qkv = (float*)d_ws;
        qkv_gemm<<<dim3(N3_ / TN, M_ / TM), dim3(256), 0, stream>>>(query, W, bias, qkv);
        attn16<<<dim3(M_ / 16), dim3(256), 0, stream>>>(qkv, mask, out);
    } else {
        fused_qkv_attn<<<dim3(M_ / 16), dim3(256), 0, stream>>>(query, mask, W, bias, out);
    }
}